// HPWL_38620345926233
// MI455X (gfx1250) — compile-verified
//
#include <hip/hip_runtime.h>

typedef __attribute__((ext_vector_type(2))) float v2f;
typedef __attribute__((ext_vector_type(8))) float v8f;

// ---- order-preserving float<->uint encoding (monotonic for all finite floats) ----
__device__ __forceinline__ unsigned ordEnc(float f) {
    unsigned u = __float_as_uint(f);
    return (u & 0x80000000u) ? ~u : (u | 0x80000000u);
}
__device__ __forceinline__ float ordDec(unsigned e) {
    return (e & 0x80000000u) ? __uint_as_float(e & 0x7FFFFFFFu)
                             : __uint_as_float(~e);
}

// ---- wave32 sum reduction via V_WMMA_F32_16X16X4_F32 ----
// A: 16x4 f32, per-lane 2 VGPRs. Lane L (0-15): A[L][0]=a.x, A[L][1]=a.y;
// lane L (16-31): A[L-16][2]=a.x, A[L-16][3]=a.y.  With B = all-ones:
//   D[m][n] = sum_k A[m][k] = s_m + s_{m+16}   (B layout irrelevant).
// D: 16x16 f32 in 8 VGPRs; VGPR r holds rows r (lanes 0-15) and r+8 (lanes 16-31).
// Summing d[0..7] in-lane gives rows 0-7 (lanes<16) / rows 8-15 (lanes>=16);
// one shfl_xor(16) completes the 32-lane sum on every lane.
__device__ __forceinline__ float waveReduceWMMA(float s) {
    v2f a; a[0] = s;    a[1] = 0.0f;
    v2f b; b[0] = 1.0f; b[1] = 1.0f;
    v8f c = {};
    v8f d = __builtin_amdgcn_wmma_f32_16x16x4_f32(
        /*neg_a=*/false, a, /*neg_b=*/false, b,
        /*c_mod=*/(short)0, c, /*reuse_a=*/false, /*reuse_b=*/false);
    float u = ((d[0] + d[1]) + (d[2] + d[3])) + ((d[4] + d[5]) + (d[6] + d[7]));
    u += __shfl_xor(u, 16, 32);
    return u;
}

// ---- kernel 1: init per-net encoded min/max ----
__global__ void hpwl_init(unsigned* __restrict__ xmax, unsigned* __restrict__ xmin,
                          unsigned* __restrict__ ymax, unsigned* __restrict__ ymin,
                          int num_nets) {
    int i = blockIdx.x * blockDim.x + threadIdx.x;
    if (i < num_nets) {
        xmax[i] = 0u;           ymax[i] = 0u;            // enc lower bound
        xmin[i] = 0xFFFFFFFFu;  ymin[i] = 0xFFFFFFFFu;   // enc upper bound
    }
}

// ---- kernel 2: scatter pins -> per-net bbox (u32 atomics, L2-resident) ----
__global__ void hpwl_scatter(const float* __restrict__ pos,
                             const int* __restrict__ pin2net,
                             unsigned* __restrict__ xmax, unsigned* __restrict__ xmin,
                             unsigned* __restrict__ ymax, unsigned* __restrict__ ymin,
                             int num_pins) {
    int i = blockIdx.x * blockDim.x + threadIdx.x;
    if (i >= num_pins) return;
    float x = pos[i];
    float y = pos[num_pins + i];
    int   net = pin2net[i];
    unsigned ex = ordEnc(x), ey = ordEnc(y);
    atomicMax(&xmax[net], ex);
    atomicMin(&xmin[net], ex);
    atomicMax(&ymax[net], ey);
    atomicMin(&ymin[net], ey);
}

// ---- kernel 3: per-net HPWL, weighted, fixed-shape deterministic partial sums ----
__global__ void hpwl_net_reduce(const unsigned* __restrict__ xmax, const unsigned* __restrict__ xmin,
                                const unsigned* __restrict__ ymax, const unsigned* __restrict__ ymin,
                                const float* __restrict__ weights,
                                const unsigned char* __restrict__ mask,
                                float* __restrict__ partials, int num_nets) {
    float acc = 0.0f;
    int stride = gridDim.x * blockDim.x;
    for (int n = blockIdx.x * blockDim.x + threadIdx.x; n < num_nets; n += stride) {
        unsigned exa = xmax[n], exi = xmin[n];
        if (mask[n] && (exa >= exi)) {   // empty nets: exa=0 < exi=0xFFFFFFFF
            float hx = ordDec(exa) - ordDec(exi);
            float hy = ordDec(ymax[n]) - ordDec(ymin[n]);
            acc += weights[n] * (hx + hy);
        }
    }
    // all 256 threads reach here (EXEC all-ones for WMMA)
    float wsum = waveReduceWMMA(acc);
    __shared__ float lds[8];
    int lane = threadIdx.x & 31;
    int wv   = threadIdx.x >> 5;
    if (lane == 0) lds[wv] = wsum;
    __syncthreads();
    if (threadIdx.x == 0) {
        float t = 0.0f;
        int nw = blockDim.x >> 5;
        for (int k = 0; k < nw; ++k) t += lds[k];
        partials[blockIdx.x] = t;
    }
}

// ---- kernel 4: final deterministic reduction of partials -> scalar ----
__global__ void hpwl_final(const float* __restrict__ partials, int n,
                           float* __restrict__ out) {
    float acc = 0.0f;
    for (int i = threadIdx.x; i < n; i += blockDim.x) acc += partials[i];
    float wsum = waveReduceWMMA(acc);
    __shared__ float lds[8];
    int lane = threadIdx.x & 31;
    int wv   = threadIdx.x >> 5;
    if (lane == 0) lds[wv] = wsum;
    __syncthreads();
    if (threadIdx.x == 0) {
        float t = 0.0f;
        int nw = blockDim.x >> 5;
        for (int k = 0; k < nw; ++k) t += lds[k];
        out[0] = t * (1.0f / 1000.0f);
    }
}

extern "C" void kernel_launch(void* const* d_in, const int* in_sizes, int n_in,
                              void* d_out, int out_size, void* d_ws, size_t ws_size,
                              hipStream_t stream) {
    const float*         pos      = (const float*)d_in[0];
    const int*           pin2net  = (const int*)d_in[1];
    const float*         weights  = (const float*)d_in[2];
    const unsigned char* mask     = (const unsigned char*)d_in[3];
    // d_in[4] = num_nets scalar (device); use host-side sizes instead:
    const int num_pins = in_sizes[1];
    const int num_nets = in_sizes[2];

    // workspace layout: 4 encoded bbox arrays + block partials
    unsigned* xmax = (unsigned*)d_ws;
    unsigned* xmin = xmax + (size_t)num_nets;
    unsigned* ymax = xmin + (size_t)num_nets;
    unsigned* ymin = ymax + (size_t)num_nets;
    float* partials = (float*)(ymin + (size_t)num_nets);

    const int BLK = 256;
    const int G_REDUCE = 3072;   // fixed shape -> deterministic partial assignment

    hpwl_init<<<(num_nets + BLK - 1) / BLK, BLK, 0, stream>>>(xmax, xmin, ymax, ymin, num_nets);
    hpwl_scatter<<<(num_pins + BLK - 1) / BLK, BLK, 0, stream>>>(pos, pin2net,
                                                                 xmax, xmin, ymax, ymin, num_pins);
    hpwl_net_reduce<<<G_REDUCE, BLK, 0, stream>>>(xmax, xmin, ymax, ymin,
                                                  weights, mask, partials, num_nets);
    hpwl_final<<<1, BLK, 0, stream>>>(partials, G_REDUCE, (float*)d_out);
}